// TransformerPredictor_56667798503473
// MI455X (gfx1250) — compile-verified
//
#include <hip/hip_runtime.h>
#include <hip/hip_bf16.h>

typedef float v2f __attribute__((ext_vector_type(2)));
typedef float v8f __attribute__((ext_vector_type(8)));
typedef unsigned int v4u __attribute__((ext_vector_type(4)));
typedef int v4i __attribute__((ext_vector_type(4)));
typedef int v8i __attribute__((ext_vector_type(8)));

#define WMMA_F32(a, b, c) \
    __builtin_amdgcn_wmma_f32_16x16x4_f32(false, (a), false, (b), (short)0, (c), false, false)

// ---------------------------------------------------------------------------
// Constants from the reference
// ---------------------------------------------------------------------------
enum : int {
    N_OBJ    = 20480,
    N_REL    = 65536,
    HID      = 512,
    POOL     = 4096,
    N_RELCLS = 51,
    N_OBJCLS = 151,
    HID2     = 2 * HID,          // 1024
    MT       = 32,               // relations per workgroup in fused kernel
    PROW     = 1028,             // P stride: 1028%64==4 -> b64 A reads conflict-free
    VROW     = 20,               // V scratch stride: 20m%64 multiples of 4 -> conflict-free
    UROW     = 66,               // U tile LDS pitch (TDM pads 64 DWORDs by 2)
    // LDS layout (floats)
    OFF_P    = 0,
    OFF_VS   = MT * PROW,                    // 32896
    OFF_UT   = OFF_VS + 8 * 16 * VROW,       // 35456
    SMEM_F   = OFF_UT + 2 * 32 * UROW        // 39680 floats = 158720 B
};

// B-operand load for the N=51 (padded to 64) weight matrices, row-major [K][51]
__device__ __forceinline__ v2f ldb51(const float* __restrict__ W, int krow, int n) {
    v2f b;
    b.x = (n < N_RELCLS) ? W[(size_t)krow * N_RELCLS + n] : 0.0f;
    b.y = (n < N_RELCLS) ? W[(size_t)(krow + 1) * N_RELCLS + n] : 0.0f;
    return b;
}

// ---------------------------------------------------------------------------
// Tensor Data Mover: DMA a 32-row x 64-col f32 tile of U into LDS at
// lds_byte_off with a row pitch of 66 floats (pad 2 DWORDs every 64).
// Hand-built D# per CDNA5 ISA 8.3/8.4; 2D tensor, groups 2/3 zero.
// This toolchain's builtin is the 6-arg (clang-23 / therock-10.0) form.
// ---------------------------------------------------------------------------
__device__ __forceinline__ void tdm_load_U_tile(const float* tile_gaddr,
                                                unsigned lds_byte_off) {
    unsigned long long ga = (unsigned long long)(size_t)tile_gaddr;
    v4u g0;
    g0.x = 1u;                                                // count=1 (valid user D#)
    g0.y = lds_byte_off;                                      // LDS destination
    g0.z = (unsigned)(ga & 0xFFFFFFFFu);                      // global_addr[31:0]
    g0.w = (unsigned)((ga >> 32) & 0x01FFFFFFu) | (2u << 30); // addr[56:32] | type=2
    v8i g1;
    g1[0] = (2 << 16)                                         // data_size = 4B
          | (1 << 20)                                         // pad_enable
          | (5 << 22)                                         // pad_interval: 64 DWORDs
          | (1 << 25);                                        // pad_amount: 2 DWORDs
    g1[1] = (int)(((unsigned)POOL & 0xFFFFu) << 16);          // tensor_dim0 lo16
    g1[2] = (int)((((unsigned)POOL >> 16) & 0xFFFFu)          // tensor_dim0 hi16
                  | (((unsigned)N_REL & 0xFFFFu) << 16));     // tensor_dim1 lo16
    g1[3] = (int)((((unsigned)N_REL >> 16) & 0xFFFFu)         // tensor_dim1 hi16
                  | (64u << 16));                             // tile_dim0 = 64
    g1[4] = 32;                                               // tile_dim1 = 32
    g1[5] = POOL;                                             // tensor_dim0_stride
    g1[6] = 0;
    g1[7] = 0;
    v4i z4 = {0, 0, 0, 0};
    v8i z8 = {0, 0, 0, 0, 0, 0, 0, 0};
    __builtin_amdgcn_tensor_load_to_lds(g0, g1, z4, z4, z8, 0);
}

// ---------------------------------------------------------------------------
// Kernel 1: edge_rep = edge_ctx @ post_emb_w + post_emb_b   -> d_ws
// ---------------------------------------------------------------------------
__global__ __launch_bounds__(256) void k_post_emb(const float* __restrict__ A,
                                                  const float* __restrict__ W,
                                                  const float* __restrict__ bias,
                                                  float* __restrict__ out) {
    const int lane = threadIdx.x & 31;
    const int wave = threadIdx.x >> 5;
    const int ncl  = lane & 15;
    const int hi   = lane >> 4;
    const int kb   = hi * 2;
    const int m0   = (blockIdx.y * 8 + wave) * 16;
    const int n0   = blockIdx.x * 16;

    const float* arow = A + (size_t)(m0 + ncl) * HID;

    v8f acc = {};
    for (int k = 0; k < HID; k += 4) {
        v2f a = *(const v2f*)(arow + k + kb);
        v2f b;
        b.x = W[(size_t)(k + kb) * HID2 + n0 + ncl];
        b.y = W[(size_t)(k + kb + 1) * HID2 + n0 + ncl];
        acc = WMMA_F32(a, b, acc);
    }

    const float bv = bias[n0 + ncl];
#pragma unroll
    for (int i = 0; i < 8; ++i) {
        out[(size_t)(m0 + hi * 8 + i) * HID2 + n0 + ncl] = acc[i] + bv;
    }
}

// ---------------------------------------------------------------------------
// Kernel 2: fused gather -> (P@Wcat + bcat) * U -> @Wrel  (+ P@Wctx)
//           with TDM double-buffered union tiles in LDS.
// ---------------------------------------------------------------------------
__global__ __launch_bounds__(256) void k_fused(const float* __restrict__ edge_rep, // [N_OBJ][1024]
                                               const float* __restrict__ U,        // [N_REL][4096]
                                               const int* __restrict__ pair_idx,   // [N_REL][2]
                                               const int* __restrict__ obj_preds,  // [N_OBJ]
                                               const float* __restrict__ Wcat,     // [1024][4096]
                                               const float* __restrict__ bcat,     // [4096]
                                               const float* __restrict__ Wrel,     // [4096][51]
                                               const float* __restrict__ brel,     // [51]
                                               const float* __restrict__ Wctx,     // [1024][51]
                                               const float* __restrict__ bctx,     // [51]
                                               const float* __restrict__ freq,     // [151*151][51]
                                               float* __restrict__ out) {          // [N_REL][51]
    extern __shared__ float smem[];
    float* P  = smem + OFF_P;                 // MT x PROW   (prod_rep tile)
    float* Vs = smem + OFF_VS;                // 8 x 16 x VROW (per-wave V scratch)
    float* Ut = smem + OFF_UT;                // 2 x 32 x UROW (TDM union tiles)

    const int tid  = threadIdx.x;
    const int lane = tid & 31;
    const int wave = tid >> 5;
    const int ncl  = lane & 15;
    const int hi   = lane >> 4;
    const int kb   = hi * 2;
    const int rel0 = blockIdx.x * MT;
    const int msub = wave & 1;                // which 16-row half of the 32-rel tile
    const int nq   = wave >> 1;               // 0..3: quarter of each 64-wide pool chunk

    // Kick off the first union tile DMA before anything else (TDM ignores EXEC;
    // issue from wave 0 only so exactly one DMA per block is in flight).
    const unsigned ut_base = (unsigned)(size_t)(void*)Ut;
    if (wave == 0)
        tdm_load_U_tile(U + (size_t)rel0 * POOL, ut_base);

    // Gather prod_rep tile: P[r][c] = edge_rep[pair(c<512 ? 0 : 1)][c]
    for (int e = tid; e < MT * (HID2 / 4); e += 256) {
        int r = e >> 8;
        int c = (e & 255) * 4;
        int src = pair_idx[2 * (rel0 + r) + (c >= HID ? 1 : 0)];
        float4 v = *(const float4*)(edge_rep + (size_t)src * HID2 + c);
        *(float4*)(P + r * PROW + c) = v;
    }
    __syncthreads();

    const float* Pa = P + (msub * 16 + ncl) * PROW;

    v8f acc0 = {}, acc1 = {}, acc2 = {}, acc3 = {};   // out N-tiles (51 valid of 64)

    // ---- ctx path: acc += P[msub] @ Wctx ; K=1024 split 4 ways by nq ----
    {
        const int k0 = nq * 256;
        for (int k = k0; k < k0 + 256; k += 4) {
            v2f a  = *(const v2f*)(Pa + k + kb);
            int kr = k + kb;
            acc0 = WMMA_F32(a, ldb51(Wctx, kr, ncl), acc0);
            acc1 = WMMA_F32(a, ldb51(Wctx, kr, 16 + ncl), acc1);
            acc2 = WMMA_F32(a, ldb51(Wctx, kr, 32 + ncl), acc2);
            acc3 = WMMA_F32(a, ldb51(Wctx, kr, 48 + ncl), acc3);
        }
    }

    // ---- main fused loop over pool columns, 64 at a time ----
    float* vs = Vs + wave * 16 * VROW;
    for (int c0 = 0; c0 < POOL; c0 += 64) {
        const int buf = (c0 >> 6) & 1;
        const float* ut = Ut + buf * (32 * UROW);

        // Hand off the TDM tile for this chunk, then prefetch the next one
        // into the other buffer (its previous readers are past the barrier).
        if (wave == 0) __builtin_amdgcn_s_wait_tensorcnt(0);
        __syncthreads();
        if (wave == 0 && c0 + 64 < POOL)
            tdm_load_U_tile(U + (size_t)rel0 * POOL + (c0 + 64),
                            ut_base + (unsigned)((buf ^ 1) * 32 * UROW * 4));

        const int cw = c0 + nq * 16;          // this wave's 16 pool columns

        // G = P(16x1024) @ Wcat[:, cw..cw+15] ; two interleaved WMMA chains
        v8f ga = {}, gb = {};
        for (int k = 0; k < HID2; k += 8) {
            v2f a1 = *(const v2f*)(Pa + k + kb);
            v2f b1;
            b1.x = Wcat[(size_t)(k + kb) * POOL + cw + ncl];
            b1.y = Wcat[(size_t)(k + kb + 1) * POOL + cw + ncl];
            ga = WMMA_F32(a1, b1, ga);
            v2f a2 = *(const v2f*)(Pa + k + 4 + kb);
            v2f b2;
            b2.x = Wcat[(size_t)(k + 4 + kb) * POOL + cw + ncl];
            b2.y = Wcat[(size_t)(k + 5 + kb) * POOL + cw + ncl];
            gb = WMMA_F32(a2, b2, gb);
        }

        // V = (G + bcat) * U(tile in LDS)  -> re-layout through padded scratch
        const float gbias = bcat[cw + ncl];
#pragma unroll
        for (int i = 0; i < 8; ++i) {
            int   ml = hi * 8 + i;            // local M row of this 16x16 tile
            float u  = ut[(msub * 16 + ml) * UROW + nq * 16 + ncl];
            vs[ml * VROW + ncl] = (ga[i] + gb[i] + gbias) * u;
        }

        // acc += V(16x16) @ Wrel[cw..cw+15, 0:51]
        for (int kk = 0; kk < 16; kk += 4) {
            v2f a  = *(const v2f*)(vs + ncl * VROW + kk + kb);
            int kr = cw + kk + kb;
            acc0 = WMMA_F32(a, ldb51(Wrel, kr, ncl), acc0);
            acc1 = WMMA_F32(a, ldb51(Wrel, kr, 16 + ncl), acc1);
            acc2 = WMMA_F32(a, ldb51(Wrel, kr, 32 + ncl), acc2);
            acc3 = WMMA_F32(a, ldb51(Wrel, kr, 48 + ncl), acc3);
        }
    }

    // ---- cross-wave reduction: reuse Vs/Ut LDS region (dead after the loop) ----
    __syncthreads();
    float* ACCs = Vs;                         // MT x 64, aliases Vs/Ut
    for (int i = tid; i < MT * 64; i += 256) ACCs[i] = 0.0f;
    __syncthreads();
#pragma unroll
    for (int i = 0; i < 8; ++i) {
        int m = msub * 16 + hi * 8 + i;
        atomicAdd(ACCs + m * 64 + ncl,      acc0[i]);   // ds_add_f32
        atomicAdd(ACCs + m * 64 + 16 + ncl, acc1[i]);
        atomicAdd(ACCs + m * 64 + 32 + ncl, acc2[i]);
        atomicAdd(ACCs + m * 64 + 48 + ncl, acc3[i]);
    }
    __syncthreads();

    // ---- final: + rel_b + ctx_b + freq_bias[pair_lbl] ----
    for (int e = tid; e < MT * N_RELCLS; e += 256) {
        int r = e / N_RELCLS, n = e - r * N_RELCLS;
        int rel = rel0 + r;
        int p0  = pair_idx[2 * rel];
        int p1  = pair_idx[2 * rel + 1];
        int lbl = obj_preds[p0] * N_OBJCLS + obj_preds[p1];
        out[(size_t)rel * N_RELCLS + n] =
            ACCs[r * 64 + n] + brel[n] + bctx[n] + freq[(size_t)lbl * N_RELCLS + n];
    }
}

// ---------------------------------------------------------------------------
extern "C" void kernel_launch(void* const* d_in, const int* in_sizes, int n_in,
                              void* d_out, int out_size, void* d_ws, size_t ws_size,
                              hipStream_t stream) {
    (void)in_sizes; (void)n_in; (void)out_size; (void)ws_size;

    const float* edge_ctx  = (const float*)d_in[0];
    const float* U         = (const float*)d_in[1];
    const int*   pair_idx  = (const int*)d_in[2];
    const int*   obj_preds = (const int*)d_in[3];
    const float* Wemb      = (const float*)d_in[4];
    const float* bemb      = (const float*)d_in[5];
    const float* Wcat      = (const float*)d_in[6];
    const float* bcat      = (const float*)d_in[7];
    const float* Wrel      = (const float*)d_in[8];
    const float* brel      = (const float*)d_in[9];
    const float* Wctx      = (const float*)d_in[10];
    const float* bctx      = (const float*)d_in[11];
    const float* freq      = (const float*)d_in[12];
    float*       out       = (float*)d_out;
    float*       edge_rep  = (float*)d_ws;   // N_OBJ x 1024 f32 (80 MB)

    // Kernel 1: edge representations
    dim3 g1(HID2 / 16, N_OBJ / (16 * 8));    // 64 x 160 blocks
    k_post_emb<<<g1, 256, 0, stream>>>(edge_ctx, Wemb, bemb, edge_rep);

    // Kernel 2: fused relation scoring
    size_t smem = (size_t)SMEM_F * sizeof(float);               // ~155 KB
    (void)hipFuncSetAttribute(reinterpret_cast<const void*>(k_fused),
                              hipFuncAttributeMaxDynamicSharedMemorySize, (int)smem);
    k_fused<<<N_REL / MT, 256, smem, stream>>>(edge_rep, U, pair_idx, obj_preds,
                                               Wcat, bcat, Wrel, brel, Wctx, bctx,
                                               freq, out);
}